// YoloFacade_36816459661564
// MI455X (gfx1250) — compile-verified
//
#include <hip/hip_runtime.h>
#include <math.h>

#define B_IMG 64
#define NPRED 25200
#define NTOP  1024
#define NCAND 2048
#define NDET  300
#define NCLS  15
#define NO    20
#define CONF  0.25f
#define IOUT  0.45f

typedef __attribute__((ext_vector_type(2))) float v2f;
typedef __attribute__((ext_vector_type(8))) float v8f;

// fast sigmoid: v_exp_f32 + v_rcp_f32 (no IEEE divide expansion); ~1ulp rcp is
// far below the 0.25-threshold decision granularity of this pipeline.
__device__ __forceinline__ float sigf(float x) {
    return __builtin_amdgcn_rcpf(1.0f + __expf(-x));
}

// wave32 broadcast via LDS-permute unit (no LDS memory touched)
__device__ __forceinline__ float bcast_f(float v, int srcLane) {
    return __int_as_float(__builtin_amdgcn_ds_bpermute(srcLane << 2, __float_as_int(v)));
}
__device__ __forceinline__ unsigned bcast_u(unsigned v, int srcLane) {
    return (unsigned)__builtin_amdgcn_ds_bpermute(srcLane << 2, (int)v);
}

// ---------------------------------------------------------------------------
// K1: fused decode + objectness/class scoring.  Streams 129MB once (BW-bound).
// Writes per-anchor: score (0 if below thresholds), class id, xyxy box.
// ---------------------------------------------------------------------------
__global__ void __launch_bounds__(256) k_decode(
        const float* __restrict__ r0, const float* __restrict__ r1,
        const float* __restrict__ r2, const float* __restrict__ strd,
        const float* __restrict__ ag,
        float* __restrict__ score, float* __restrict__ cls,
        float4* __restrict__ box)
{
    int gid = blockIdx.x * blockDim.x + threadIdx.x;
    if (gid >= B_IMG * NPRED) return;
    int img = gid / NPRED;
    int k   = gid - img * NPRED;

    const float* raw; int lvl, nx, loc;
    if (k < 19200)      { lvl = 0; nx = 80; loc = k;         raw = r0; }
    else if (k < 24000) { lvl = 1; nx = 40; loc = k - 19200; raw = r1; }
    else                { lvl = 2; nx = 20; loc = k - 24000; raw = r2; }

    int cells = nx * nx;
    int a   = loc / cells;
    int rem = loc - a * cells;
    int y   = rem / nx;
    int x   = rem - y * nx;

    const float4* p4 = reinterpret_cast<const float4*>(
        raw + ((((size_t)img * 3 + a) * cells) + rem) * NO);
    float4 q0 = p4[0];            // tx ty tw th
    float4 q1 = p4[1];            // obj c0 c1 c2
    float4 q2 = p4[2];            // c3..c6
    float4 q3 = p4[3];            // c7..c10
    float4 q4 = p4[4];            // c11..c14

    float s  = strd[lvl];
    float ax = ag[lvl * 6 + a * 2 + 0];
    float ay = ag[lvl * 6 + a * 2 + 1];

    float sx = sigf(q0.x), sy = sigf(q0.y), sw = sigf(q0.z), sh = sigf(q0.w);
    float cx = (sx * 2.0f - 0.5f + (float)x) * s;
    float cy = (sy * 2.0f - 0.5f + (float)y) * s;
    float w  = (sw * 2.0f) * (sw * 2.0f) * ax;
    float h  = (sh * 2.0f) * (sh * 2.0f) * ay;

    float obj = sigf(q1.x);
    float cr[NCLS] = { q1.y, q1.z, q1.w, q2.x, q2.y, q2.z, q2.w,
                       q3.x, q3.y, q3.z, q3.w, q4.x, q4.y, q4.z, q4.w };
    float mraw = cr[0]; int mc = 0;
    #pragma unroll
    for (int c = 1; c < NCLS; ++c) { if (cr[c] > mraw) { mraw = cr[c]; mc = c; } }

    float best = obj * sigf(mraw);                  // sigmoid monotone -> argmax of raw
    float sc   = (obj > CONF && best > CONF) ? best : 0.0f;

    score[gid] = sc;
    cls[gid]   = (float)mc;
    box[gid]   = make_float4(cx - 0.5f * w, cy - 0.5f * h, cx + 0.5f * w, cy + 0.5f * h);
}

// ---------------------------------------------------------------------------
// K2: per-image top-1024 via exact 2048-bin radix select + LDS bitonic sort.
// Score in (0.25,1) -> float bits>>13 is an order-preserving 2048-value key.
// ---------------------------------------------------------------------------
__global__ void __launch_bounds__(256) k_topk(
        const float* __restrict__ score, const float* __restrict__ cls,
        const float4* __restrict__ box,
        float* __restrict__ tsc, float4* __restrict__ tbox, float* __restrict__ tcls)
{
    __shared__ unsigned hist[2048];
    __shared__ float cs[NCAND];
    __shared__ int   ci[NCAND];
    __shared__ int   sT, sAbove, cntA, cntT;

    int img = blockIdx.x;
    int tid = threadIdx.x;
    const float* s = score + (size_t)img * NPRED;

    for (int i = tid; i < 2048; i += 256) hist[i] = 0u;
    for (int i = tid; i < NCAND; i += 256) { cs[i] = 0.0f; ci[i] = 0x7FFFFFFF; }
    if (tid == 0) { cntA = 0; cntT = 0; }
    __syncthreads();

    for (int i = tid; i < NPRED; i += 256) {
        float v = s[i];
        if (v > CONF) {
            int b = (int)(__float_as_uint(v) >> 13) - 0x1F400;
            b = b < 0 ? 0 : (b > 2047 ? 2047 : b);
            atomicAdd(&hist[b], 1u);
        }
    }
    __syncthreads();

    if (tid == 0) {
        int tot = 0, T = -1, above = 0;
        for (int b = 2047; b >= 0; --b) {
            unsigned hcnt = hist[b];
            if (tot + (int)hcnt >= NTOP) { T = b; above = tot; break; }
            tot += (int)hcnt;
        }
        sT = T; sAbove = above;
    }
    __syncthreads();
    int T = sT, above = sAbove;

    for (int i = tid; i < NPRED; i += 256) {
        float v = s[i];
        if (v > CONF) {
            int b = (int)(__float_as_uint(v) >> 13) - 0x1F400;
            b = b < 0 ? 0 : (b > 2047 ? 2047 : b);
            int pos = -1;
            if (b > T)       pos = atomicAdd(&cntA, 1);
            else if (b == T) pos = above + atomicAdd(&cntT, 1);
            if (pos >= 0 && pos < NCAND) { cs[pos] = v; ci[pos] = i; }
        }
    }
    __syncthreads();

    // bitonic sort, key = (score desc, index asc)
    for (int k = 2; k <= NCAND; k <<= 1) {
        for (int j = k >> 1; j > 0; j >>= 1) {
            for (int t = tid; t < NCAND; t += 256) {
                int ixj = t ^ j;
                if (ixj > t) {
                    float a0 = cs[t], a1 = cs[ixj];
                    int   i0 = ci[t], i1 = ci[ixj];
                    bool up = ((t & k) == 0);
                    bool oo = (a1 > a0) || (a1 == a0 && i1 < i0); // ixj ranks before t
                    if (up == oo) { cs[t] = a1; cs[ixj] = a0; ci[t] = i1; ci[ixj] = i0; }
                }
            }
            __syncthreads();
        }
    }

    for (int p = tid; p < NTOP; p += 256) {
        float v  = cs[p];
        int  idx = (v > 0.0f) ? ci[p] : 0;
        tsc [(size_t)img * NTOP + p] = v;
        tbox[(size_t)img * NTOP + p] = box[(size_t)img * NPRED + idx];
        tcls[(size_t)img * NTOP + p] = cls[(size_t)img * NPRED + idx];
    }
}

// ---------------------------------------------------------------------------
// K3: suppression bitmask.  One block/image, one wave per 16x16 IoU tile.
// Union denominator area[i]+area[j] for the whole tile comes from a single
// V_WMMA_F32_16X16X4_F32 rank-2 outer product; the intersection min/max math
// lands on the documented C/D lane layout (VGPR r -> rows r,r+8).
// The IoU threshold test is divide-free: union > 0 always, so
// iou > t  <=>  inter > t * (sumArea - inter + eps)   (one FMA + one cmp).
// ---------------------------------------------------------------------------
__global__ void __launch_bounds__(256) k_mask(const float4* __restrict__ tbox,
                                              unsigned* __restrict__ mask)
{
    int img  = blockIdx.x;
    int tid  = threadIdx.x;
    int wave = tid >> 5, lane = tid & 31;
    unsigned* m = mask + (size_t)img * NTOP * 32;

    for (int i = tid; i < NTOP * 32; i += 256) m[i] = 0u;
    __syncthreads();

    const float4* bx = tbox + (size_t)img * NTOP;

    for (int rt = 0; rt < 64; ++rt) {
        float4 rb = bx[rt * 16 + (lane & 15)];               // row boxes (dup in hi half)
        float areaR = (rb.z - rb.x) * (rb.w - rb.y);
        for (int ct = rt + wave; ct < 64; ct += 8) {         // upper triangle only
            float4 cb = bx[ct * 16 + (lane & 15)];           // col boxes
            float areaC = (cb.z - cb.x) * (cb.w - cb.y);

            v2f a, b;
            if (lane < 16) { a[0] = areaR; a[1] = 1.0f; b[0] = 1.0f; b[1] = areaC; }
            else           { a[0] = 0.0f;  a[1] = 0.0f; b[0] = 0.0f; b[1] = 0.0f;  }
            v8f c = {0.f, 0.f, 0.f, 0.f, 0.f, 0.f, 0.f, 0.f};
            // D[m][n] = areaR[m]*1 + 1*areaC[n]  (16x16 pairwise-sum tile)
            v8f d = __builtin_amdgcn_wmma_f32_16x16x4_f32(
                        false, a, false, b, (short)0, c, false, false);

            #pragma unroll
            for (int r = 0; r < 8; ++r) {
                int mrow = r + ((lane >> 4) << 3);           // lanes 0-15: r, 16-31: r+8
                float rx1 = bcast_f(rb.x, mrow);
                float ry1 = bcast_f(rb.y, mrow);
                float rx2 = bcast_f(rb.z, mrow);
                float ry2 = bcast_f(rb.w, mrow);
                float iw = fmaxf(fminf(rx2, cb.z) - fmaxf(rx1, cb.x), 0.0f);
                float ih = fmaxf(fminf(ry2, cb.w) - fmaxf(ry1, cb.y), 0.0f);
                float inter = iw * ih;
                // union = d[r] - inter + eps  (> 0 always): divide-free compare
                float uni = d[r] - inter + 1e-7f;
                unsigned bal = __builtin_amdgcn_ballot_w32(inter > IOUT * uni);
                if (lane == 0) {
                    int word = ct >> 1, sh = (ct & 1) * 16;
                    atomicOr(&m[(rt * 16 + r)     * 32 + word], (bal & 0xFFFFu) << sh);
                    atomicOr(&m[(rt * 16 + r + 8) * 32 + word], (bal >> 16)     << sh);
                }
            }
        }
    }
}

// ---------------------------------------------------------------------------
// K4: serial greedy suppression (single wave32 bit-sweep) + stable compaction
// into the first <=300 output rows [x1 y1 x2 y2 score cls]; rest zeroed.
// ---------------------------------------------------------------------------
__global__ void __launch_bounds__(32) k_nms_out(
        const float* __restrict__ tsc, const float4* __restrict__ tbox,
        const float* __restrict__ tcls, const unsigned* __restrict__ mask,
        float* __restrict__ out)
{
    int img  = blockIdx.x;
    int lane = threadIdx.x;
    const float*    s = tsc  + (size_t)img * NTOP;
    const unsigned* m = mask + (size_t)img * NTOP * 32;

    unsigned keep = 0u;                          // lane owns columns [lane*32, lane*32+31]
    for (int b = 0; b < 32; ++b)
        if (s[lane * 32 + b] > CONF) keep |= (1u << b);

    for (int i = 0; i < NTOP; ++i) {
        int wi = i >> 5;
        unsigned w = bcast_u(keep, wi);
        if ((w >> (i & 31)) & 1u) {              // row i still kept
            unsigned row = m[i * 32 + lane];     // coalesced 128B row of iou bits
            int bi = i & 31;
            unsigned g;
            if (lane > wi)      g = 0xFFFFFFFFu;
            else if (lane < wi) g = 0u;
            else                g = (bi == 31) ? 0u : (0xFFFFFFFFu << (bi + 1));
            keep &= ~(row & g);
        }
    }

    float* o = out + (size_t)img * NDET * 6;
    int base = 0;
    for (int wdx = 0; wdx < 32 && base < NDET; ++wdx) {
        unsigned w = bcast_u(keep, wdx);
        if (w) {
            if ((w >> lane) & 1u) {
                unsigned below = (lane == 0) ? 0u : (w & (0xFFFFFFFFu >> (32 - lane)));
                int rank = base + __popc(below);
                if (rank < NDET) {
                    int i = wdx * 32 + lane;
                    float4 bx = tbox[(size_t)img * NTOP + i];
                    float* q = o + rank * 6;
                    q[0] = bx.x; q[1] = bx.y; q[2] = bx.z; q[3] = bx.w;
                    q[4] = s[i]; q[5] = tcls[(size_t)img * NTOP + i];
                }
            }
            base += __popc(w);
        }
    }
    int nz = base < NDET ? base : NDET;
    for (int f = nz * 6 + lane; f < NDET * 6; f += 32) o[f] = 0.0f;
}

// ---------------------------------------------------------------------------
extern "C" void kernel_launch(void* const* d_in, const int* in_sizes, int n_in,
                              void* d_out, int out_size, void* d_ws, size_t ws_size,
                              hipStream_t stream)
{
    const float* r0 = (const float*)d_in[0];
    const float* r1 = (const float*)d_in[1];
    const float* r2 = (const float*)d_in[2];
    const float* st = (const float*)d_in[3];
    const float* ag = (const float*)d_in[4];

    char* ws = (char*)d_ws;
    size_t off = 0;
    float*    score = (float*)   (ws + off); off += (size_t)B_IMG * NPRED * 4;
    float*    cls   = (float*)   (ws + off); off += (size_t)B_IMG * NPRED * 4;
    float4*   box   = (float4*)  (ws + off); off += (size_t)B_IMG * NPRED * 16;
    float*    tsc   = (float*)   (ws + off); off += (size_t)B_IMG * NTOP * 4;
    float4*   tbox  = (float4*)  (ws + off); off += (size_t)B_IMG * NTOP * 16;
    float*    tcls  = (float*)   (ws + off); off += (size_t)B_IMG * NTOP * 4;
    unsigned* msk   = (unsigned*)(ws + off); off += (size_t)B_IMG * NTOP * 32 * 4;

    int total = B_IMG * NPRED;
    k_decode<<<(total + 255) / 256, 256, 0, stream>>>(r0, r1, r2, st, ag, score, cls, box);
    k_topk  <<<B_IMG, 256, 0, stream>>>(score, cls, box, tsc, tbox, tcls);
    k_mask  <<<B_IMG, 256, 0, stream>>>(tbox, msk);
    k_nms_out<<<B_IMG, 32, 0, stream>>>(tsc, tbox, tcls, msk, (float*)d_out);
}